// SubSampler_69793218560118
// MI455X (gfx1250) — compile-verified
//
#include <hip/hip_runtime.h>
#include <hip/hip_bf16.h>
#include <math.h>

// ---------------------------------------------------------------------------
// SubSampler for MI455X (gfx1250, wave32, WMMA).
//
// Pipeline: zeroAcc -> sortTopK (bitonic in 256KB LDS) -> phaseA (online
// softmax stats per particle) -> phaseB (WMMA f32 16x16x4 accumulation of
// att @ [x0, x1, 1] with exp recomputed on the fly) -> finalize.
// ---------------------------------------------------------------------------

#define NPART   32768
#define MCENT   2048
#define TAUF    1e-3f
#define INVTAU  (1.0f / TAUF)
#define PSPLIT  16           // particle splits in phase B
#define CHUNK   512          // particles staged in LDS per chunk (8 KB)

typedef float v2f __attribute__((ext_vector_type(2)));
typedef float v8f __attribute__((ext_vector_type(8)));

// ---------------------------------------------------------------- zero accum
__global__ void zeroAcc(float* __restrict__ acc) {
    int i = blockIdx.x * 256 + threadIdx.x;
    if (i < MCENT * 3) acc[i] = 0.0f;
}

// ------------------------------------------------- top-k via bitonic in LDS
// key = (float_bits(w) << 32) | (0xFFFFFFFF - idx); w >= 0 so float bits are
// monotone in value.  Descending sort => ties resolve to the smaller index
// first, matching jax.lax.top_k.
__global__ __launch_bounds__(1024) void sortTopK(const float* __restrict__ x,
                                                 const float* __restrict__ w,
                                                 float4* __restrict__ cent) {
    extern __shared__ unsigned long long key[];   // 32768 * 8B = 256 KB LDS
    for (int i = threadIdx.x; i < NPART; i += 1024) {
        unsigned wb = __float_as_uint(w[i]);
        key[i] = ((unsigned long long)wb << 32) |
                 (unsigned long long)(0xFFFFFFFFu - (unsigned)i);
    }
    __syncthreads();

    for (int k = 2; k <= NPART; k <<= 1) {
        for (int j = k >> 1; j > 0; j >>= 1) {
            for (int i = threadIdx.x; i < NPART; i += 1024) {
                int l = i ^ j;
                if (l > i) {
                    unsigned long long a = key[i], b = key[l];
                    bool desc = ((i & k) == 0);
                    bool doSwap = desc ? (a < b) : (a > b);
                    if (doSwap) { key[i] = b; key[l] = a; }
                }
            }
            __syncthreads();
        }
    }

    // Emit centroid records {y0, y1, -|y|^2/tau, 0} for the top-2048.
    for (int m = threadIdx.x; m < MCENT; m += 1024) {
        unsigned idx = 0xFFFFFFFFu - (unsigned)(key[m] & 0xFFFFFFFFull);
        float y0 = x[2 * idx + 0];
        float y1 = x[2 * idx + 1];
        cent[m] = make_float4(y0, y1, -(y0 * y0 + y1 * y1) * INVTAU, 0.0f);
    }
}

// ------------------------------------------------ phase A: softmax statistics
// For particle n:  u(m) = min( (2/tau) x.y_m - |x|^2/tau - |y_m|^2/tau, 0 )
// Online max/sum over m.  Store {q0, q1, nxs, ladj} with
// ladj = ln(w_n) - ln(Z_n) - rowmax_n  so phase B needs ONE exp per pair:
//   att[m,n]*w_n = exp( min(q0*ym0 + q1*ym1 + nxs + g_m, 0) + ladj ).
__global__ __launch_bounds__(256) void phaseA(const float* __restrict__ x,
                                              const float* __restrict__ w,
                                              const float4* __restrict__ cent,
                                              float4* __restrict__ part4) {
    __shared__ float cy0[MCENT], cy1[MCENT], cg[MCENT];   // 24 KB
    for (int i = threadIdx.x; i < MCENT; i += 256) {
        float4 c = cent[i];
        cy0[i] = c.x; cy1[i] = c.y; cg[i] = c.z;
    }
    __syncthreads();

    int n = blockIdx.x * 256 + threadIdx.x;
    float x0 = x[2 * n + 0], x1 = x[2 * n + 1], wn = w[n];
    float q0 = x0 * (2.0f * INVTAU);
    float q1 = x1 * (2.0f * INVTAU);
    float nxs = -(x0 * x0 + x1 * x1) * INVTAU;

    float mx = -INFINITY, s = 0.0f;
    for (int m = 0; m < MCENT; ++m) {
        float u = fminf(fmaf(q0, cy0[m], fmaf(q1, cy1[m], nxs)) + cg[m], 0.0f);
        float nm = fmaxf(mx, u);
        s = s * __expf(mx - nm) + __expf(u - nm);
        mx = nm;
    }
    float ladj = __logf(wn) - __logf(s) - mx;
    part4[n] = make_float4(q0, q1, nxs, ladj);
}

// -------------------------------------------- phase B: WMMA accumulation
// Each wave owns 16 centroids (A rows).  Per step: 4 particles (K=4).
// A lane layout (32-bit A 16x4): lane l -> row m = l%16; VGPR0 holds K =
// (l<16 ? 0 : 2), VGPR1 holds K+1.  B (4x16): VGPR0 row K=(l<16?0:2) at
// col l%16, VGPR1 row K+1.  B cols: 0 -> x0, 1 -> x1, 2 -> 1, else 0.
// C (16x16 f32, 8 VGPRs): VGPR r, lanes0-15 -> (M=r, N=lane),
// lanes16-31 -> (M=r+8, N=lane-16).  Only N=0..2 are harvested.
__global__ __launch_bounds__(256) void phaseB(const float4* __restrict__ part4,
                                              const float4* __restrict__ cent,
                                              float* __restrict__ acc) {
    __shared__ float4 lp[CHUNK];                         // 8 KB
    const int lane  = threadIdx.x & 31;
    const int wave  = threadIdx.x >> 5;                  // 0..7
    const int col   = lane & 15;
    const int hi    = lane >> 4;                         // 0 or 1
    const int mBase = (blockIdx.x * 8 + wave) * 16;      // 16 centroids / wave

    float4 cm = cent[mBase + col];
    const float ym0 = cm.x, ym1 = cm.y, gm = cm.z;

    v8f c = {};
    const int pBase = blockIdx.y * (NPART / PSPLIT);
    for (int cs = 0; cs < NPART / PSPLIT; cs += CHUNK) {
        __syncthreads();
        lp[threadIdx.x]       = part4[pBase + cs + threadIdx.x];
        lp[threadIdx.x + 256] = part4[pBase + cs + threadIdx.x + 256];
        if (cs + CHUNK < NPART / PSPLIT)
            __builtin_prefetch(&part4[pBase + cs + CHUNK + threadIdx.x], 0, 1);
        __syncthreads();

        for (int k0 = 0; k0 < CHUNK; k0 += 4) {
            float4 pA = lp[k0 + 2 * hi];         // particle K   (this lane)
            float4 pB = lp[k0 + 2 * hi + 1];     // particle K+1 (this lane)

            float uA = fminf(fmaf(pA.x, ym0, fmaf(pA.y, ym1, pA.z)) + gm, 0.0f);
            float uB = fminf(fmaf(pB.x, ym0, fmaf(pB.y, ym1, pB.z)) + gm, 0.0f);

            v2f A;
            A[0] = __expf(uA + pA.w);            // att[m,K]   * w_K
            A[1] = __expf(uB + pB.w);            // att[m,K+1] * w_{K+1}

            v2f B;
            B[0] = (col == 0) ? pA.x * (0.5f * TAUF)
                 : (col == 1) ? pA.y * (0.5f * TAUF)
                 : (col == 2) ? 1.0f : 0.0f;
            B[1] = (col == 0) ? pB.x * (0.5f * TAUF)
                 : (col == 1) ? pB.y * (0.5f * TAUF)
                 : (col == 2) ? 1.0f : 0.0f;

            c = __builtin_amdgcn_wmma_f32_16x16x4_f32(
                    /*neg_a=*/false, A, /*neg_b=*/false, B,
                    /*c_mod=*/(short)0, c, /*reuse_a=*/false, /*reuse_b=*/false);
        }
    }

    if (col < 3) {
#pragma unroll
        for (int r = 0; r < 8; ++r)
            atomicAdd(&acc[(mBase + r + hi * 8) * 3 + col], c[r]);
    }
}

// ------------------------------------------------------------------ finalize
__global__ void finalize(const float* __restrict__ acc, float* __restrict__ out) {
    int m = blockIdx.x * 256 + threadIdx.x;
    if (m < MCENT) {
        float n0 = acc[3 * m + 0];
        float n1 = acc[3 * m + 1];
        float d  = acc[3 * m + 2];
        out[2 * m + 0] = n0 / d;
        out[2 * m + 1] = n1 / d;
        out[2 * MCENT + m] = d;          // v == den
    }
}

// ---------------------------------------------------------------------------
extern "C" void kernel_launch(void* const* d_in, const int* in_sizes, int n_in,
                              void* d_out, int out_size, void* d_ws, size_t ws_size,
                              hipStream_t stream) {
    const float* x = (const float*)d_in[0];   // [32768, 2]
    const float* w = (const float*)d_in[1];   // [32768]
    float* out = (float*)d_out;               // 2048*2 (y) + 2048 (v)

    char* ws = (char*)d_ws;
    // u64 sort keys live in sortTopK's dynamic LDS, not in ws
    // [0, 512K): per-particle float4 (written in phaseA, after keys retire)
    float4* part4 = (float4*)(ws + 0);
    float4* cent  = (float4*)(ws + 524288);          // 2048 * 16B = 32 KB
    float*  acc   = (float*) (ws + 524288 + 32768);  // 2048 * 3 floats = 24 KB

    zeroAcc<<<dim3((MCENT * 3 + 255) / 256), dim3(256), 0, stream>>>(acc);
    sortTopK<<<dim3(1), dim3(1024), NPART * sizeof(unsigned long long), stream>>>(x, w, cent);
    phaseA<<<dim3(NPART / 256), dim3(256), 0, stream>>>(x, w, cent, part4);
    phaseB<<<dim3(MCENT / (8 * 16), PSPLIT), dim3(256), 0, stream>>>(part4, cent, acc);
    finalize<<<dim3((MCENT + 255) / 256), dim3(256), 0, stream>>>(acc, out);
}